// GATLayer_549755814580
// MI455X (gfx1250) — compile-verified
//
#include <hip/hip_runtime.h>
#include <hip/hip_bf16.h>

#define NN 50000
#define EE 800000
#define KF 256       // IN_F
#define HH 4
#define CC 64
#define HC 256       // H*C
#define OC 64        // final out channels
#define TOT (EE + NN)
#define NEG_SLOPE 0.2f
#define BN_EPS 1e-5f

typedef float v2f __attribute__((ext_vector_type(2)));
typedef float v8f __attribute__((ext_vector_type(8)));

// Monotonic float<->uint encoding for atomicMax-based segment max.
__device__ __forceinline__ unsigned enc_f32(float f) {
  unsigned b = __float_as_uint(f);
  return b ^ ((unsigned)((int)b >> 31) | 0x80000000u);
}
__device__ __forceinline__ float dec_f32(unsigned e) {
  unsigned b = (e & 0x80000000u) ? (e ^ 0x80000000u) : ~e;
  return __uint_as_float(b);
}

// ---------------------------------------------------------------- init
__global__ void k_init(float* __restrict__ agg, float* __restrict__ denom,
                       unsigned* __restrict__ menc) {
  size_t i = (size_t)blockIdx.x * blockDim.x + threadIdx.x;
  if (i < (size_t)NN * HC) agg[i] = 0.f;
  if (i < (size_t)NN * HH) { denom[i] = 0.f; menc[i] = 0u; }
}

// ------------------------------------------------- GEMM1: xp = x @ W^T
// grid (3125, 2), block 256 (8 waves). Each block: one 16-row M-tile,
// each wave: one 16x16 output tile (obase = blockIdx.y*128 + wave*16).
__global__ __launch_bounds__(256) void k_gemm_proj(const float* __restrict__ x,
                                                   const float* __restrict__ W,
                                                   float* __restrict__ xp) {
  __shared__ float shx[16][KF + 4];   // +4 pad: 16B aligned, bank-conflict free
  const int tid  = threadIdx.x;
  const int row0 = blockIdx.x * 16;

  // cooperative stage of the 16x256 x tile (16 floats per thread)
  {
    const int r  = tid >> 4;
    const int c0 = (tid & 15) * 16;
    const float* src = &x[(size_t)(row0 + r) * KF + c0];
#pragma unroll
    for (int i = 0; i < 16; i += 4) {
      float4 v = *reinterpret_cast<const float4*>(src + i);
      *reinterpret_cast<float4*>(&shx[r][c0 + i]) = v;
    }
  }
  __syncthreads();

  const int wave  = tid >> 5;
  const int lane  = tid & 31;
  const int lrow  = lane & 15;
  const int kh    = (lane >> 4) << 1;           // 0 or 2
  const int obase = blockIdx.y * 128 + wave * 16;
  const float* wrow = &W[(size_t)(obase + lrow) * KF];

  __builtin_prefetch(wrow, 0, 1);
  __builtin_prefetch(wrow + 128, 0, 1);

  v8f acc = {0.f, 0.f, 0.f, 0.f, 0.f, 0.f, 0.f, 0.f};
#pragma unroll 8
  for (int k = 0; k < KF; k += 4) {
    v2f a = *reinterpret_cast<const v2f*>(&shx[lrow][k + kh]);
    v2f b = *reinterpret_cast<const v2f*>(wrow + k + kh);
    acc = __builtin_amdgcn_wmma_f32_16x16x4_f32(false, a, false, b,
                                                (short)0, acc, false, false);
  }
  const int rbase = row0 + ((lane >> 4) << 3);   // rows 0-7 or 8-15
#pragma unroll
  for (int i = 0; i < 8; ++i)
    xp[(size_t)(rbase + i) * HC + obase + lrow] = acc[i];
}

// --------------------------------- per-node logits: a_src/a_dst [N,H]
// one wave per node; lane l handles elements o = l*8..l*8+7 (head = l/8)
__global__ __launch_bounds__(256) void k_logits(const float* __restrict__ xp,
                                                const float* __restrict__ att_src,
                                                const float* __restrict__ att_dst,
                                                float* __restrict__ a_src,
                                                float* __restrict__ a_dst) {
  const int wid = (blockIdx.x * blockDim.x + threadIdx.x) >> 5;
  if (wid >= NN) return;
  const int lane = threadIdx.x & 31;
  const int o0 = lane * 8;
  const float* row = &xp[(size_t)wid * HC + o0];
  float ps = 0.f, pd = 0.f;
#pragma unroll
  for (int j = 0; j < 8; ++j) {
    float v = row[j];
    ps += v * att_src[o0 + j];
    pd += v * att_dst[o0 + j];
  }
  // reduce across the 8 lanes of each head group
#pragma unroll
  for (int off = 1; off < 8; off <<= 1) {
    ps += __shfl_xor(ps, off, 32);
    pd += __shfl_xor(pd, off, 32);
  }
  if ((lane & 7) == 0) {
    const int h = lane >> 3;
    a_src[wid * HH + h] = ps;
    a_dst[wid * HH + h] = pd;
  }
}

// --------------------------------------------- segment max over edges
__global__ void k_edge_max(const int* __restrict__ ei,
                           const float* __restrict__ asrc,
                           const float* __restrict__ adst,
                           unsigned* __restrict__ menc) {
  const int i = blockIdx.x * blockDim.x + threadIdx.x;
  if (i >= TOT) return;
  int s, d;
  if (i < EE) { s = ei[i]; d = ei[EE + i]; } else { s = d = i - EE; }
#pragma unroll
  for (int h = 0; h < HH; ++h) {
    float e = asrc[s * HH + h] + adst[d * HH + h];
    e = (e >= 0.f) ? e : NEG_SLOPE * e;
    atomicMax(&menc[d * HH + h], enc_f32(e));
  }
}

// -------------------------------------------- segment softmax denom
__global__ void k_edge_denom(const int* __restrict__ ei,
                             const float* __restrict__ asrc,
                             const float* __restrict__ adst,
                             const unsigned* __restrict__ menc,
                             float* __restrict__ denom) {
  const int i = blockIdx.x * blockDim.x + threadIdx.x;
  if (i >= TOT) return;
  int s, d;
  if (i < EE) { s = ei[i]; d = ei[EE + i]; } else { s = d = i - EE; }
#pragma unroll
  for (int h = 0; h < HH; ++h) {
    float e = asrc[s * HH + h] + adst[d * HH + h];
    e = (e >= 0.f) ? e : NEG_SLOPE * e;
    atomicAdd(&denom[d * HH + h], __expf(e - dec_f32(menc[d * HH + h])));
  }
}

// ------------------------------- weighted message scatter-aggregation
// one wave per edge; lane l moves elements o = l*8..l*8+7 (head = l/8)
__global__ __launch_bounds__(256) void k_aggregate(const int* __restrict__ ei,
                                                   const float* __restrict__ xp,
                                                   const float* __restrict__ asrc,
                                                   const float* __restrict__ adst,
                                                   const unsigned* __restrict__ menc,
                                                   const float* __restrict__ denom,
                                                   float* __restrict__ agg) {
  const int wid = blockIdx.x * 8 + (threadIdx.x >> 5);
  if (wid >= TOT) return;
  const int lane = threadIdx.x & 31;
  int s, d;
  if (wid < EE) { s = ei[wid]; d = ei[EE + wid]; } else { s = d = wid - EE; }
  const int h = lane >> 3;
  float e = asrc[s * HH + h] + adst[d * HH + h];
  e = (e >= 0.f) ? e : NEG_SLOPE * e;
  const float alpha = __expf(e - dec_f32(menc[d * HH + h])) /
                      (denom[d * HH + h] + 1e-16f);
  const int o0 = lane * 8;
  const float* xs = &xp[(size_t)s * HC + o0];
  float* dsto = &agg[(size_t)d * HC + o0];
#pragma unroll
  for (int j = 0; j < 8; ++j) atomicAdd(&dsto[j], alpha * xs[j]);
}

// ----- final: out = relu(BN(agg + gat_bias)) @ lin_W^T + lin_b (WMMA)
// grid 3125, block 128 (4 waves). Block stages BN'd 16x256 act tile in
// LDS; wave w computes output columns w*16..w*16+15.
__global__ __launch_bounds__(128) void k_gemm_out(const float* __restrict__ agg,
                                                  const float* __restrict__ gat_bias,
                                                  const float* __restrict__ gamma,
                                                  const float* __restrict__ beta,
                                                  const float* __restrict__ mean,
                                                  const float* __restrict__ var,
                                                  const float* __restrict__ linW,
                                                  const float* __restrict__ linb,
                                                  float* __restrict__ out) {
  __shared__ float sha[16][HC + 4];
  const int tid  = threadIdx.x;
  const int row0 = blockIdx.x * 16;
  {
    const int r  = tid >> 3;          // 0..15
    const int c0 = (tid & 7) * 32;    // 0..224
    const float* src = &agg[(size_t)(row0 + r) * HC + c0];
#pragma unroll
    for (int i = 0; i < 32; i += 4) {
      const int c = c0 + i;
      float4 v  = *reinterpret_cast<const float4*>(src + i);
      float4 gb = *reinterpret_cast<const float4*>(&gat_bias[c]);
      float4 mu = *reinterpret_cast<const float4*>(&mean[c]);
      float4 vr = *reinterpret_cast<const float4*>(&var[c]);
      float4 gm = *reinterpret_cast<const float4*>(&gamma[c]);
      float4 bt = *reinterpret_cast<const float4*>(&beta[c]);
      float4 o;
      o.x = fmaxf((v.x + gb.x - mu.x) * rsqrtf(vr.x + BN_EPS) * gm.x + bt.x, 0.f);
      o.y = fmaxf((v.y + gb.y - mu.y) * rsqrtf(vr.y + BN_EPS) * gm.y + bt.y, 0.f);
      o.z = fmaxf((v.z + gb.z - mu.z) * rsqrtf(vr.z + BN_EPS) * gm.z + bt.z, 0.f);
      o.w = fmaxf((v.w + gb.w - mu.w) * rsqrtf(vr.w + BN_EPS) * gm.w + bt.w, 0.f);
      *reinterpret_cast<float4*>(&sha[r][c]) = o;
    }
  }
  __syncthreads();

  const int wave  = tid >> 5;
  const int lane  = tid & 31;
  const int lrow  = lane & 15;
  const int kh    = (lane >> 4) << 1;
  const int obase = wave * 16;
  const float* wrow = &linW[(size_t)(obase + lrow) * HC];
  __builtin_prefetch(wrow, 0, 1);

  v8f acc = {0.f, 0.f, 0.f, 0.f, 0.f, 0.f, 0.f, 0.f};
#pragma unroll 8
  for (int k = 0; k < HC; k += 4) {
    v2f a = *reinterpret_cast<const v2f*>(&sha[lrow][k + kh]);
    v2f b = *reinterpret_cast<const v2f*>(wrow + k + kh);
    acc = __builtin_amdgcn_wmma_f32_16x16x4_f32(false, a, false, b,
                                                (short)0, acc, false, false);
  }
  const float bias = linb[obase + lrow];
  const int rbase = row0 + ((lane >> 4) << 3);
#pragma unroll
  for (int i = 0; i < 8; ++i)
    out[(size_t)(rbase + i) * OC + obase + lrow] = acc[i] + bias;
}

// --------------------------------------------------------------------
extern "C" void kernel_launch(void* const* d_in, const int* in_sizes, int n_in,
                              void* d_out, int out_size, void* d_ws, size_t ws_size,
                              hipStream_t stream) {
  (void)in_sizes; (void)n_in; (void)out_size; (void)ws_size;
  const float* x        = (const float*)d_in[0];
  const int*   ei       = (const int*)d_in[1];
  const float* W        = (const float*)d_in[2];
  const float* att_src  = (const float*)d_in[3];
  const float* att_dst  = (const float*)d_in[4];
  const float* gat_bias = (const float*)d_in[5];
  const float* bn_gamma = (const float*)d_in[6];
  const float* bn_beta  = (const float*)d_in[7];
  const float* bn_mean  = (const float*)d_in[8];
  const float* bn_var   = (const float*)d_in[9];
  const float* lin_W    = (const float*)d_in[10];
  const float* lin_b    = (const float*)d_in[11];
  float* out = (float*)d_out;

  float* ws    = (float*)d_ws;
  float* agg   = ws;                                   // N*HC
  float* xp    = agg  + (size_t)NN * HC;               // N*HC
  float* asrc  = xp   + (size_t)NN * HC;               // N*H
  float* adst  = asrc + (size_t)NN * HH;               // N*H
  float* denom = adst + (size_t)NN * HH;               // N*H
  unsigned* menc = (unsigned*)(denom + (size_t)NN * HH); // N*H

  {
    const size_t n = (size_t)NN * HC;
    k_init<<<(unsigned)((n + 255) / 256), 256, 0, stream>>>(agg, denom, menc);
  }
  k_gemm_proj<<<dim3(NN / 16, 2), 256, 0, stream>>>(x, W, xp);
  k_logits<<<(NN * 32 + 255) / 256, 256, 0, stream>>>(xp, att_src, att_dst, asrc, adst);
  k_edge_max<<<(TOT + 255) / 256, 256, 0, stream>>>(ei, asrc, adst, menc);
  k_edge_denom<<<(TOT + 255) / 256, 256, 0, stream>>>(ei, asrc, adst, menc, denom);
  k_aggregate<<<(TOT + 7) / 8, 256, 0, stream>>>(ei, xp, asrc, adst, menc, denom, agg);
  k_gemm_out<<<NN / 16, 128, 0, stream>>>(agg, gat_bias, bn_gamma, bn_beta,
                                          bn_mean, bn_var, lin_W, lin_b, out);
}